// GraceCell_26611617366178
// MI455X (gfx1250) — compile-verified
//
#include <hip/hip_runtime.h>
#include <math.h>

// ---------------------------------------------------------------------------
// GRACE-style 2x 3-layer GAT on gfx1250 (MI455X).
//   - GEMMs (x @ W) via V_WMMA_F32_16X16X4_F32 (fp32, exact vs reference;
//     GEMM is ~8 GFLOP total, negligible next to the edge scatter phase).
//   - Edge softmax via atomic segment-max / segment-sum (L2-resident: the
//     25.6MB node-feature arrays fit in the 192MB L2).
//   - Aggregation: one wave32 per edge, global_prefetch on h[src].
// ---------------------------------------------------------------------------

typedef float v2f __attribute__((ext_vector_type(2)));
typedef float v8f __attribute__((ext_vector_type(8)));

__device__ __forceinline__ void atomicMaxF32(float* addr, float val) {
  // IEEE-754 monotonicity trick; addr must be initialized to -inf.
  if (val >= 0.0f) {
    atomicMax((int*)addr, __float_as_int(val));
  } else {
    atomicMin((unsigned int*)addr, __float_as_uint(val));
  }
}

// H[nrows x Dout] = X[nrows x K] @ W[K x Dout]; one wave per 16x16 tile.
// fp32 WMMA 16x16x4. Requires nrows%16==0, K%4==0, Dout%16==0 (holds here).
__global__ void gat_gemm_wmma(const float* __restrict__ X,
                              const float* __restrict__ W,
                              float* __restrict__ H,
                              int nrows, int K, int Dout) {
  const int wavesPerBlk = blockDim.x >> 5;
  const int wid  = blockIdx.x * wavesPerBlk + ((int)threadIdx.x >> 5);
  const int lane = threadIdx.x & 31;
  const int ntc   = Dout >> 4;
  const int tiles = (nrows >> 4) * ntc;
  if (wid >= tiles) return;                 // wave-uniform: EXEC stays all-1s
  const int tm = wid / ntc, tn = wid - tm * ntc;
  const int row0 = tm << 4, col0 = tn << 4;
  const int half = lane >> 4, l16 = lane & 15;

  // A 16x4 layout: lanes 0-15 -> K={k,k+1}, lanes 16-31 -> K={k+2,k+3}, M=l16
  // B 4x16 layout: VGPR v, lanes 0-15 -> K=k+v, lanes 16-31 -> K=k+2+v, N=l16
  const float* __restrict__ xr = X + (size_t)(row0 + l16) * K;
  const float* __restrict__ wc = W + col0 + l16;

  v8f acc = {};
  for (int k = 0; k < K; k += 4) {
    const int ks = k + (half << 1);
    v2f a, b;
    a.x = xr[ks];
    a.y = xr[ks + 1];
    b.x = wc[(size_t)ks * Dout];
    b.y = wc[(size_t)(ks + 1) * Dout];
    acc = __builtin_amdgcn_wmma_f32_16x16x4_f32(false, a, false, b, (short)0,
                                                acc, false, false);
  }
  // C/D 16x16 f32: VGPR r -> row r (lanes 0-15) / row r+8 (lanes 16-31)
  float* hp = H + (size_t)(row0 + (half << 3)) * Dout + col0 + l16;
#pragma unroll
  for (int r = 0; r < 8; ++r) hp[(size_t)r * Dout] = acc[r];
}

// als[n] = h[n,:] . asrc ; ald[n] = h[n,:] . adst   (one wave32 per node)
__global__ void gat_attn_coef(const float* __restrict__ H,
                              const float* __restrict__ asrc,
                              const float* __restrict__ adst,
                              float* __restrict__ als, float* __restrict__ ald,
                              int n, int D) {
  const int wv   = blockIdx.x * (blockDim.x >> 5) + ((int)threadIdx.x >> 5);
  const int lane = threadIdx.x & 31;
  if (wv >= n) return;
  const float* h = H + (size_t)wv * D;
  float ss = 0.f, sd = 0.f;
  for (int f = lane; f < D; f += 32) {
    float v = h[f];
    ss = fmaf(v, asrc[f], ss);
    sd = fmaf(v, adst[f], sd);
  }
#pragma unroll
  for (int off = 16; off > 0; off >>= 1) {
    ss += __shfl_down(ss, off, 32);
    sd += __shfl_down(sd, off, 32);
  }
  if (lane == 0) { als[wv] = ss; ald[wv] = sd; }
}

__global__ void gat_init_ms(float* __restrict__ m, float* __restrict__ s, int n) {
  int i = blockIdx.x * blockDim.x + threadIdx.x;
  if (i < n) { m[i] = __int_as_float(0xFF800000); s[i] = 0.0f; }
}

__global__ void gat_init_out(float* __restrict__ out, const float* __restrict__ b,
                             int n, int D) {
  int i = blockIdx.x * blockDim.x + threadIdx.x;
  if (i < n * D) out[i] = b[i & (D - 1)];   // D is 128 or 64 (power of two)
}

// pass 1: logits = leaky_relu(als[src]+ald[dst]); segment-max into m[dst]
__global__ void gat_edge_logit_max(const long long* __restrict__ ei, int E, int n,
                                   const float* __restrict__ als,
                                   const float* __restrict__ ald,
                                   float* __restrict__ lg, float* __restrict__ m) {
  int e = blockIdx.x * blockDim.x + threadIdx.x;
  if (e >= E + n) return;
  int s, d;
  if (e < E) { s = (int)ei[e]; d = (int)ei[E + e]; } else { s = d = e - E; }
  float x = als[s] + ald[d];
  x = (x > 0.0f) ? x : 0.2f * x;
  lg[e] = x;
  atomicMaxF32(m + d, x);
}

// pass 2: p = exp(logit - m[dst]); segment-sum into s[dst]
__global__ void gat_edge_exp_sum(const long long* __restrict__ ei, int E, int n,
                                 const float* __restrict__ m,
                                 float* __restrict__ lg, float* __restrict__ s) {
  int e = blockIdx.x * blockDim.x + threadIdx.x;
  if (e >= E + n) return;
  int d = (e < E) ? (int)ei[E + e] : (e - E);
  float p = expf(lg[e] - m[d]);
  lg[e] = p;
  atomicAdd(s + d, p);
}

// pass 3: out[dst,:] += (p/s[dst]) * h[src,:]   (one wave32 per edge)
__global__ void gat_aggregate(const long long* __restrict__ ei, int E, int n, int D,
                              const float* __restrict__ H,
                              const float* __restrict__ lg,
                              const float* __restrict__ sden,
                              float* __restrict__ out) {
  const int wv   = blockIdx.x * (blockDim.x >> 5) + ((int)threadIdx.x >> 5);
  const int lane = threadIdx.x & 31;
  if (wv >= E + n) return;
  int s, d;
  if (wv < E) { s = (int)ei[wv]; d = (int)ei[E + wv]; } else { s = d = wv - E; }
  const float* __restrict__ hs = H + (size_t)s * D;
  __builtin_prefetch(hs + lane * 4, 0, 0);       // global_prefetch_b8 on gather row
  float alpha = lg[wv] / sden[d];
  float* op = out + (size_t)d * D;
  for (int f = lane; f < D; f += 32) atomicAdd(op + f, alpha * hs[f]);
}

// ---------------------------------------------------------------------------

static inline int cdiv(int a, int b) { return (a + b - 1) / b; }

static void run_gat_layer(const float* xin, int din, int dout,
                          const float* W, const float* asrc, const float* adst,
                          const float* bias, const long long* ei, int E, int n,
                          float* h, float* als, float* ald, float* m, float* s,
                          float* lg, float* out, hipStream_t stream) {
  const int TB = 256, WPB = TB / 32;
  const int tiles = (n / 16) * (dout / 16);
  const int tot = E + n;
  gat_gemm_wmma<<<cdiv(tiles, WPB), TB, 0, stream>>>(xin, W, h, n, din, dout);
  gat_attn_coef<<<cdiv(n, WPB), TB, 0, stream>>>(h, asrc, adst, als, ald, n, dout);
  gat_init_ms<<<cdiv(n, TB), TB, 0, stream>>>(m, s, n);
  gat_init_out<<<cdiv(n * dout, TB), TB, 0, stream>>>(out, bias, n, dout);
  gat_edge_logit_max<<<cdiv(tot, TB), TB, 0, stream>>>(ei, E, n, als, ald, lg, m);
  gat_edge_exp_sum<<<cdiv(tot, TB), TB, 0, stream>>>(ei, E, n, m, lg, s);
  gat_aggregate<<<cdiv(tot, WPB), TB, 0, stream>>>(ei, E, n, dout, h, lg, s, out);
}

extern "C" void kernel_launch(void* const* d_in, const int* in_sizes, int n_in,
                              void* d_out, int out_size, void* d_ws, size_t ws_size,
                              hipStream_t stream) {
  (void)n_in; (void)out_size; (void)ws_size;

  const float* x        = (const float*)d_in[0];
  const float* augx     = (const float*)d_in[1];
  const long long* ei1  = (const long long*)d_in[2];
  const long long* ei2  = (const long long*)d_in[3];
  const float *W[6], *as[6], *ad[6], *bs[6];
  for (int i = 0; i < 6; ++i) {
    W[i]  = (const float*)d_in[4 + 4 * i];
    as[i] = (const float*)d_in[5 + 4 * i];
    ad[i] = (const float*)d_in[6 + 4 * i];
    bs[i] = (const float*)d_in[7 + 4 * i];
  }

  const int dhid = in_sizes[7];            // |b1| = 128
  const int dout = in_sizes[15];           // |b3| = 64
  const int din  = in_sizes[4] / dhid;     // W1 rows = 128
  const int n    = in_sizes[0] / din;      // 50000
  const int E1   = in_sizes[2] / 2;        // 800000
  const int E2   = in_sizes[3] / 2;
  const int Emax = (E1 > E2) ? E1 : E2;

  // workspace carve-up (~81 MB)
  float* ws  = (float*)d_ws;
  float* h   = ws;  ws += (size_t)n * dhid;    // GEMM output per layer
  float* p0  = ws;  ws += (size_t)n * dhid;    // ping
  float* p1  = ws;  ws += (size_t)n * dhid;    // pong
  float* als = ws;  ws += n;
  float* ald = ws;  ws += n;
  float* m   = ws;  ws += n;
  float* s   = ws;  ws += n;
  float* lg  = ws;  ws += (size_t)Emax + n;

  float* out1 = (float*)d_out;
  float* out2 = out1 + (size_t)n * dout;

  // path 1: x -> GAT1 -> GAT2 -> GAT3
  run_gat_layer(x,  din,  dhid, W[0], as[0], ad[0], bs[0], ei1, E1, n, h, als, ald, m, s, lg, p0,   stream);
  run_gat_layer(p0, dhid, dhid, W[1], as[1], ad[1], bs[1], ei1, E1, n, h, als, ald, m, s, lg, p1,   stream);
  run_gat_layer(p1, dhid, dout, W[2], as[2], ad[2], bs[2], ei1, E1, n, h, als, ald, m, s, lg, out1, stream);

  // path 2: aug_x -> GAT4 -> GAT5 -> GAT6
  run_gat_layer(augx, din,  dhid, W[3], as[3], ad[3], bs[3], ei2, E2, n, h, als, ald, m, s, lg, p0,   stream);
  run_gat_layer(p0,   dhid, dhid, W[4], as[4], ad[4], bs[4], ei2, E2, n, h, als, ald, m, s, lg, p1,   stream);
  run_gat_layer(p1,   dhid, dout, W[5], as[5], ad[5], bs[5], ei2, E2, n, h, als, ald, m, s, lg, out2, stream);
}